// model_S0_5351529251078
// MI455X (gfx1250) — compile-verified
//
#include <hip/hip_runtime.h>
#include <hip/hip_bf16.h>
#include <math.h>

// ---------------------------------------------------------------------------
// MI455X (gfx1250) implementation of the DDI GNN forward pass.
//
// The AE1 fingerprint-attention branch is dead code w.r.t. the returned
// tensor and is skipped. The live path is gather/atomic bound (~3.2 GB edge
// traffic => ~140us floor at 23.3 TB/s HBM), so GEMMs stay in fp32 using
// V_WMMA_F32_16X16X4_F32 (bf16 WMMA adds nothing when HBM is the limiter).
//
// Workspace requirement: 6 * N_NODES*256 floats + softmax scratch (~250 MB).
// ---------------------------------------------------------------------------

typedef float v2f __attribute__((ext_vector_type(2)));
typedef float v8f __attribute__((ext_vector_type(8)));

// ---------------------------------------------------------------------------
// Tiled fp32 WMMA GEMM:  C[M,N] = A[M,K] (row-major) @ B[K,N] (row-major) + bias
// One wave computes a 16(M) x 64(N) tile with 4 named accumulators; the A
// fragment is loaded once per k-step and reused across the 4 N-subtiles.
// All addressing uses 32-bit element offsets from uniform bases so the
// backend can use SGPR-base + 32-bit-VGPR-offset (GVS) global addressing.
// Ragged N is handled by clamping B load columns (WMMA output columns are
// independent, so clamped columns produce garbage that is never stored) —
// the inner loop is branch-free with EXEC all-ones.
// act: 0 = bias only, 1 = bias + ELU.  Requires M % 16 == 0, K % 4 == 0.
// ---------------------------------------------------------------------------
__global__ void wmma_gemm_f32(const float* __restrict__ A,
                              const float* __restrict__ B,
                              const float* __restrict__ bias,
                              float* __restrict__ C,
                              int M, int N, int K, int act) {
  const int lane = threadIdx.x & 31;
  const int waveInBlock = threadIdx.x >> 5;
  const int nT = (N + 63) >> 6;          // 64-wide N tiles
  const int mT = M >> 4;                 // 16-tall M tiles
  const int gid = blockIdx.x * (blockDim.x >> 5) + waveInBlock;
  if (gid >= mT * nT) return;            // wave-uniform: EXEC stays all-ones

  const int tm = (gid / nT) << 4;
  const int tn = (gid % nT) << 6;

  // ISA 7.12.2 fp32 A 16x4 fragment: lanes 0-15 -> K={0,1}, lanes 16-31 -> K={2,3}
  const int fm   = lane & 15;            // M (A) / N (B) index within tile
  const int koff = (lane >> 4) << 1;     // 0 or 2

  // Clamped per-fragment B columns (branch-free ragged-N handling).
  const int c0 = tn + fm;
  const int c1 = tn + 16 + fm;
  const int c2 = tn + 32 + fm;
  const int c3 = tn + 48 + fm;
  const unsigned cl0 = (unsigned)((c0 < N) ? c0 : (N - 1));
  const unsigned cl1 = (unsigned)((c1 < N) ? c1 : (N - 1));
  const unsigned cl2 = (unsigned)((c2 < N) ? c2 : (N - 1));
  const unsigned cl3 = (unsigned)((c3 < N) ? c3 : (N - 1));

  // 32-bit running element offsets (uniform base stays in SGPRs).
  unsigned aoff = (unsigned)(tm + fm) * (unsigned)K + (unsigned)koff;
  unsigned o0 = (unsigned)koff * (unsigned)N + cl0;
  unsigned o1 = (unsigned)koff * (unsigned)N + cl1;
  unsigned o2 = (unsigned)koff * (unsigned)N + cl2;
  unsigned o3 = (unsigned)koff * (unsigned)N + cl3;
  const unsigned uN = (unsigned)N;
  const unsigned bStep = 4u * uN;

  v8f acc0 = {0.f,0.f,0.f,0.f,0.f,0.f,0.f,0.f};
  v8f acc1 = {0.f,0.f,0.f,0.f,0.f,0.f,0.f,0.f};
  v8f acc2 = {0.f,0.f,0.f,0.f,0.f,0.f,0.f,0.f};
  v8f acc3 = {0.f,0.f,0.f,0.f,0.f,0.f,0.f,0.f};

#pragma unroll 2
  for (int k0 = 0; k0 < K; k0 += 4) {
    const v2f a = *(const v2f*)(A + aoff);

    v2f b;
    b.x = B[o0]; b.y = B[o0 + uN];
    acc0 = __builtin_amdgcn_wmma_f32_16x16x4_f32(false, a, false, b,
                                                 (short)0, acc0, false, false);
    b.x = B[o1]; b.y = B[o1 + uN];
    acc1 = __builtin_amdgcn_wmma_f32_16x16x4_f32(false, a, false, b,
                                                 (short)0, acc1, false, false);
    b.x = B[o2]; b.y = B[o2 + uN];
    acc2 = __builtin_amdgcn_wmma_f32_16x16x4_f32(false, a, false, b,
                                                 (short)0, acc2, false, false);
    b.x = B[o3]; b.y = B[o3 + uN];
    acc3 = __builtin_amdgcn_wmma_f32_16x16x4_f32(false, a, false, b,
                                                 (short)0, acc3, false, false);

    aoff += 4u;
    o0 += bStep; o1 += bStep; o2 += bStep; o3 += bStep;
  }

  // C/D layout: VGPR i -> M = tm + i (lanes 0-15) or tm + i + 8 (lanes 16-31)
  const int rowBase = tm + ((lane >> 4) << 3);
  const v8f* accs[4] = {&acc0, &acc1, &acc2, &acc3};
#pragma unroll
  for (int j = 0; j < 4; ++j) {
    const int col = tn + (j << 4) + fm;
    if (col >= N) continue;              // store guard only (cold path)
    const float bv = bias ? bias[col] : 0.f;
    const v8f av = *accs[j];
    unsigned coff = (unsigned)rowBase * uN + (unsigned)col;
#pragma unroll
    for (int i = 0; i < 8; ++i) {
      float r = av[i] + bv;
      if (act == 1) r = (r > 0.f) ? r : (__expf(r) - 1.f);   // ELU
      C[coff] = r;
      coff += uN;
    }
  }
}

// ---------------------------------------------------------------------------
// Edge attention pass 1: per-(edge,head) score, exp, and per-dst denominator.
// One wave per edge; both heads reduced with shfl_xor.
// ---------------------------------------------------------------------------
__global__ void edge_scores(const float* __restrict__ Q,
                            const float* __restrict__ Kf,
                            const int* __restrict__ src,
                            const int* __restrict__ dst,
                            float* __restrict__ eexp,
                            float* __restrict__ denom,
                            int nEdges) {
  const int e = (int)(((size_t)blockIdx.x * blockDim.x + threadIdx.x) >> 5);
  const int lane = threadIdx.x & 31;
  if (e >= nEdges) return;
  const int s = src[e], d = dst[e];
  const float* q = Q + (size_t)d * 256;
  const float* k = Kf + (size_t)s * 256;
  float s0 = 0.f, s1 = 0.f;
#pragma unroll
  for (int c = 0; c < 128; c += 32) {
    s0 += q[c + lane] * k[c + lane];
    s1 += q[c + lane + 128] * k[c + lane + 128];
  }
#pragma unroll
  for (int off = 16; off > 0; off >>= 1) {
    s0 += __shfl_xor(s0, off, 32);
    s1 += __shfl_xor(s1, off, 32);
  }
  if (lane == 0) {
    const float inv = 0.08838834764831845f;   // 1/sqrt(128)
    const float e0 = __expf(s0 * inv);
    const float e1 = __expf(s1 * inv);
    eexp[(size_t)e * 2 + 0] = e0;
    eexp[(size_t)e * 2 + 1] = e1;
    atomicAdd(&denom[(size_t)d * 2 + 0], e0);
    atomicAdd(&denom[(size_t)d * 2 + 1], e1);
  }
}

// ---------------------------------------------------------------------------
// Edge attention pass 2: agg[dst] += alpha * v[src], per head.
// ---------------------------------------------------------------------------
__global__ void edge_aggregate(const float* __restrict__ V,
                               const float* __restrict__ eexp,
                               const float* __restrict__ denom,
                               const int* __restrict__ src,
                               const int* __restrict__ dst,
                               float* __restrict__ agg,
                               int nEdges) {
  const int e = (int)(((size_t)blockIdx.x * blockDim.x + threadIdx.x) >> 5);
  const int lane = threadIdx.x & 31;
  if (e >= nEdges) return;
  const int s = src[e], d = dst[e];
  const float a0 = eexp[(size_t)e * 2 + 0] / (denom[(size_t)d * 2 + 0] + 1e-16f);
  const float a1 = eexp[(size_t)e * 2 + 1] / (denom[(size_t)d * 2 + 1] + 1e-16f);
  const float* v = V + (size_t)s * 256;
  float* o = agg + (size_t)d * 256;
#pragma unroll
  for (int c = 0; c < 256; c += 32) {
    const int ch = c + lane;
    atomicAdd(&o[ch], v[ch] * ((ch < 128) ? a0 : a1));
  }
}

// h = (agg + skip)  (relu applied for conv1)
__global__ void add_skip(const float* __restrict__ agg,
                         const float* __restrict__ skip,
                         float* __restrict__ H, size_t n, int relu) {
  size_t i = (size_t)blockIdx.x * blockDim.x + threadIdx.x;
  const size_t stride = (size_t)gridDim.x * blockDim.x;
  for (; i < n; i += stride) {
    float r = agg[i] + skip[i];
    if (relu) r = fmaxf(r, 0.f);
    H[i] = r;
  }
}

// allv[b, 0:256] = h[drug1s[b]]; allv[b, 256:512] = h[drug2s[b]]
__global__ void gather_pairs(const float* __restrict__ H,
                             const int* __restrict__ d1,
                             const int* __restrict__ d2,
                             float* __restrict__ allv, int nPairs) {
  size_t i = (size_t)blockIdx.x * blockDim.x + threadIdx.x;
  const size_t total = (size_t)nPairs * 512;
  const size_t stride = (size_t)gridDim.x * blockDim.x;
  for (; i < total; i += stride) {
    const int b = (int)(i >> 9);
    const int c = (int)(i & 511);
    const int node = (c < 256) ? d1[b] : d2[b];
    allv[i] = H[(size_t)node * 256 + (c & 255)];
  }
}

__global__ void fill_zero(float* __restrict__ p, size_t n) {
  size_t i = (size_t)blockIdx.x * blockDim.x + threadIdx.x;
  const size_t stride = (size_t)gridDim.x * blockDim.x;
  for (; i < n; i += stride) p[i] = 0.f;
}

// ---------------------------------------------------------------------------
// Host launcher.
// Input flattening assumption: setup_inputs() dict order with params flattened
// recursively in insertion order:
//   0:x 1:morgen 2..23:ae1(dead) 24..31:c1{qW,qb,kW,kb,vW,vb,sW,sb}
//   32..39:c2{...} 40:W1 41:b1 42:W2 43:b2 44:edge_index 45:drug1s 46:drug2s
// ---------------------------------------------------------------------------
extern "C" void kernel_launch(void* const* d_in, const int* in_sizes, int n_in,
                              void* d_out, int out_size, void* d_ws, size_t ws_size,
                              hipStream_t stream) {
  (void)n_in; (void)out_size; (void)ws_size;

  const float* x = (const float*)d_in[0];

  const float* c1W[4] = {(const float*)d_in[24], (const float*)d_in[26],
                         (const float*)d_in[28], (const float*)d_in[30]};
  const float* c1b[4] = {(const float*)d_in[25], (const float*)d_in[27],
                         (const float*)d_in[29], (const float*)d_in[31]};
  const float* c2W[4] = {(const float*)d_in[32], (const float*)d_in[34],
                         (const float*)d_in[36], (const float*)d_in[38]};
  const float* c2b[4] = {(const float*)d_in[33], (const float*)d_in[35],
                         (const float*)d_in[37], (const float*)d_in[39]};
  const float* W1 = (const float*)d_in[40];
  const float* b1 = (const float*)d_in[41];
  const float* W2 = (const float*)d_in[42];
  const float* b2 = (const float*)d_in[43];
  const int* ei  = (const int*)d_in[44];
  const int* d1  = (const int*)d_in[45];
  const int* d2  = (const int*)d_in[46];

  const int nNodes = in_sizes[0] / 768;     // 40000
  const int nEdges = in_sizes[44] / 2;      // 400000
  const int nPairs = in_sizes[45];          // 16384
  const int* esrc = ei;
  const int* edst = ei + nEdges;

  // Workspace layout (floats)
  float* ws = (float*)d_ws;
  const size_t SZ = (size_t)nNodes * 256;
  float* bufQ   = ws;                 // also reused as allv [nPairs,512]
  float* bufK   = bufQ + SZ;          // also reused as hid  [nPairs,512]
  float* bufV   = bufK + SZ;
  float* bufS   = bufV + SZ;
  float* bufAgg = bufS + SZ;
  float* bufH   = bufAgg + SZ;
  float* denom  = bufH + SZ;                          // [nNodes, 2]
  float* eexp   = denom + (size_t)nNodes * 2;         // [nEdges, 2]

  const dim3 blk(256);
  auto gemmBlocks = [](int M, int N) {
    int tiles = (M / 16) * ((N + 63) / 64);
    return dim3((unsigned)((tiles + 7) / 8));
  };
  const dim3 edgeGrid((unsigned)((nEdges + 7) / 8));       // 8 waves/block
  const dim3 elemGrid(4096);

  float* qkvs[4] = {bufQ, bufK, bufV, bufS};

  // ---------------- conv1 : x[40000,768] -> h1 ----------------
  for (int i = 0; i < 4; ++i)
    wmma_gemm_f32<<<gemmBlocks(nNodes, 256), blk, 0, stream>>>(
        x, c1W[i], c1b[i], qkvs[i], nNodes, 256, 768, 0);
  fill_zero<<<elemGrid, blk, 0, stream>>>(denom, (size_t)nNodes * 2);
  fill_zero<<<elemGrid, blk, 0, stream>>>(bufAgg, SZ);
  edge_scores<<<edgeGrid, blk, 0, stream>>>(bufQ, bufK, esrc, edst, eexp, denom, nEdges);
  edge_aggregate<<<edgeGrid, blk, 0, stream>>>(bufV, eexp, denom, esrc, edst, bufAgg, nEdges);
  add_skip<<<elemGrid, blk, 0, stream>>>(bufAgg, bufS, bufH, SZ, /*relu=*/1);

  // ---------------- conv2 : h1[40000,256] -> h2 ----------------
  for (int i = 0; i < 4; ++i)
    wmma_gemm_f32<<<gemmBlocks(nNodes, 256), blk, 0, stream>>>(
        bufH, c2W[i], c2b[i], qkvs[i], nNodes, 256, 256, 0);
  fill_zero<<<elemGrid, blk, 0, stream>>>(denom, (size_t)nNodes * 2);
  fill_zero<<<elemGrid, blk, 0, stream>>>(bufAgg, SZ);
  edge_scores<<<edgeGrid, blk, 0, stream>>>(bufQ, bufK, esrc, edst, eexp, denom, nEdges);
  edge_aggregate<<<edgeGrid, blk, 0, stream>>>(bufV, eexp, denom, esrc, edst, bufAgg, nEdges);
  add_skip<<<elemGrid, blk, 0, stream>>>(bufAgg, bufS, bufH, SZ, /*relu=*/0);

  // ---------------- pair head ----------------
  float* allv = bufQ;   // [nPairs, 512]  (bufQ free after conv2 scores)
  float* hid  = bufK;   // [nPairs, 512]
  gather_pairs<<<elemGrid, blk, 0, stream>>>(bufH, d1, d2, allv, nPairs);
  wmma_gemm_f32<<<gemmBlocks(nPairs, 512), blk, 0, stream>>>(
      allv, W1, b1, hid, nPairs, 512, 512, /*act=ELU*/1);
  wmma_gemm_f32<<<gemmBlocks(nPairs, 65), blk, 0, stream>>>(
      hid, W2, b2, (float*)d_out, nPairs, 65, 512, 0);
}